// WeightedEulerCurve_41463614276199
// MI455X (gfx1250) — compile-verified
//
#include <hip/hip_runtime.h>

typedef __attribute__((ext_vector_type(2))) float v2f;
typedef __attribute__((ext_vector_type(8))) float v8f;

#define NBIN 65          // num_steps + 1 thresholds
#define HB   66          // padded LDS histogram row stride

__device__ __forceinline__ int wect_bucket(float h) {
    const float MINH  = -110.85125168440814f;   // -sqrt(3)*64
    const float INVDH = 0.28867513459481287f;   // 1/(2*sqrt(3))
    int s = (int)__builtin_ceilf((h - MINH) * INVDH);
    s = s < 0 ? 0 : (s > 64 ? 64 : s);
    return s;
}

__global__ void __launch_bounds__(256) wect_zero(float* gh, int n) {
    int i = blockIdx.x * 256 + threadIdx.x;
    if (i < n) gh[i] = 0.0f;
}

// grid.x = 4 dir-tiles * 64 chunks = 256 blocks; 256 threads = 8 wave32
__global__ void __launch_bounds__(256) wect_main(const float* __restrict__ X,
                                                 const float* __restrict__ dirs,
                                                 float* __restrict__ gh) {
    __shared__ float lhist[16 * HB];
    const int tid     = threadIdx.x;
    const int lane    = tid & 31;
    const int wave    = tid >> 5;
    const int dirTile = blockIdx.x >> 6;   // 0..3  -> 16 directions each
    const int chunk   = blockIdx.x & 63;   // 0..63 -> spatial slice
    const int dirBase = dirTile * 16;

    for (int q = tid; q < 16 * HB; q += 256) lhist[q] = 0.0f;
    __syncthreads();

    // ---- Build the 8 A-matrices (one per cell type), 16x4 f32 each ----
    // A layout (32-bit 16x4): lanes 0-15 (M=lane) hold K=0,1; lanes 16-31 hold K=2,3.
    const int   m  = dirBase + (lane & 15);
    const float e0 = dirs[3 * m + 0];
    const float e1 = dirs[3 * m + 1];
    const float e2 = dirs[3 * m + 2];
    const float p0 = fmaxf(e0, 0.0f), p1 = fmaxf(e1, 0.0f), p2 = fmaxf(e2, 0.0f);
    const bool  hiHalf = (lane & 16) != 0;
    const int   hiAdd  = hiHalf ? 8 : 0;

    const float off[8] = {0.0f, p0, p1, p2, p1 + p2, p0 + p2, p0 + p1, p0 + p1 + p2};
    v2f A[8];
#pragma unroll
    for (int t = 0; t < 8; ++t) {
        A[t].x = hiHalf ? e2 : e0;        // K=0 -> d0 ; K=2 -> d2
        A[t].y = hiHalf ? off[t] : e1;    // K=1 -> d1 ; K=3 -> type offset
    }

    // 16384 cell tiles total; 512 wave-slots * 32 tiles each. Uniform per wave
    // -> EXEC stays all-ones around the WMMAs.
    for (int it = 0; it < 32; ++it) {
        const int tile = (chunk * 8 + wave) * 32 + it;     // 0..16383
        const int n    = tile * 16 + (lane & 15);          // linear voxel id
        const int i = n >> 12, j = (n >> 6) & 63, l = n & 63;

        __builtin_prefetch(X + ((n + 16) & 262143), 0, 0); // global_prefetch_b8

        const int ip = i < 63 ? i + 1 : i;
        const int jp = j < 63 ? j + 1 : j;
        const int lp = l < 63 ? l + 1 : l;
        const float x000 = X[(i * 64 + j) * 64 + l];
        const float x100 = X[(ip * 64 + j) * 64 + l];
        const float x010 = X[(i * 64 + jp) * 64 + l];
        const float x001 = X[(i * 64 + j) * 64 + lp];
        const float x110 = X[(ip * 64 + jp) * 64 + l];
        const float x101 = X[(ip * 64 + j) * 64 + lp];
        const float x011 = X[(i * 64 + jp) * 64 + lp];
        const float x111 = X[(ip * 64 + jp) * 64 + lp];

        const bool mi = i < 63, mj = j < 63, ml = l < 63;
        float sval[8];
        sval[0] = x000;                                                     // vertex  +
        sval[1] = mi ? -fmaxf(x000, x100) : 0.0f;                           // edge x  -
        sval[2] = mj ? -fmaxf(x000, x010) : 0.0f;                           // edge y  -
        sval[3] = ml ? -fmaxf(x000, x001) : 0.0f;                           // edge z  -
        sval[4] = (mj && ml) ? fmaxf(fmaxf(x000, x010), fmaxf(x001, x011)) : 0.0f; // face yz +
        sval[5] = (mi && ml) ? fmaxf(fmaxf(x000, x100), fmaxf(x001, x101)) : 0.0f; // face xz +
        sval[6] = (mi && mj) ? fmaxf(fmaxf(x000, x100), fmaxf(x010, x110)) : 0.0f; // face xy +
        sval[7] = (mi && mj && ml)
                      ? -fmaxf(fmaxf(fmaxf(x000, x100), fmaxf(x010, x110)),
                               fmaxf(fmaxf(x001, x101), fmaxf(x011, x111)))
                      : 0.0f;                                               // cube    -

        // B matrix (4x16 f32): lanes 0-15 hold K=0,1 = (i, j); lanes 16-31 hold K=2,3 = (l, 1)
        v2f B;
        B.x = hiHalf ? (float)l : (float)i;
        B.y = hiHalf ? 1.0f     : (float)j;

#pragma unroll
        for (int t = 0; t < 8; ++t) {
            v8f c = {};
            // D(16 dirs x 16 cells) = A_t(16x4) * B(4x16): heights of type-t cells
            v8f d = __builtin_amdgcn_wmma_f32_16x16x4_f32(
                false, A[t], false, B, (short)0, c, false, false);
#pragma unroll
            for (int r = 0; r < 8; ++r) {
                const int s = wect_bucket(d[r]);             // first threshold >= height
                atomicAdd(&lhist[(r + hiAdd) * HB + s], sval[t]);  // ds_add_f32
            }
        }
    }
    __syncthreads();

    for (int q = tid; q < 16 * NBIN; q += 256) {
        const int mm = q / NBIN, ss = q % NBIN;
        atomicAdd(&gh[(dirBase + mm) * NBIN + ss], lhist[mm * HB + ss]); // global_atomic_add_f32
    }
}

// Per-direction prefix sum over the 65 signed bins -> out[k, s] = v - e + s - c at threshold s
__global__ void wect_finalize(const float* __restrict__ gh, float* __restrict__ out) {
    const int k = blockIdx.x;
    const int s = threadIdx.x;
    if (s < NBIN) {
        float acc = 0.0f;
        for (int q = 0; q <= s; ++q) acc += gh[k * NBIN + q];
        out[k * NBIN + s] = acc;
    }
}

extern "C" void kernel_launch(void* const* d_in, const int* in_sizes, int n_in,
                              void* d_out, int out_size, void* d_ws, size_t ws_size,
                              hipStream_t stream) {
    const float* X    = (const float*)d_in[0];   // (64,64,64) f32
    const float* dirs = (const float*)d_in[1];   // (64,3)     f32
    float*       out  = (float*)d_out;           // (64,65)    f32
    float*       gh   = (float*)d_ws;            // 64*65 f32 signed histogram (16.6 KB)

    wect_zero<<<(64 * NBIN + 255) / 256, 256, 0, stream>>>(gh, 64 * NBIN);
    wect_main<<<256, 256, 0, stream>>>(X, dirs, gh);
    wect_finalize<<<64, 96, 0, stream>>>(gh, out);
}